// Vdw_33741263078050
// MI455X (gfx1250) — compile-verified
//
#include <hip/hip_runtime.h>

// Output geometry: two tensors of shape (B=8, C=4, R=512, NALT=8), f32,
// concatenated flat in d_out:  [finalMC | finalSC]
#define B_DIM 8
#define C_DIM 4
#define R_DIM 512
#define NALT  8
#define OUT_PER (B_DIM * C_DIM * R_DIM * NALT)   // 131072 floats per tensor

// Native clang ext-vector types: required for __builtin_nontemporal_load
// (HIP_vector_type structs are rejected by the builtin).
typedef __attribute__((ext_vector_type(4))) int          v4i;
typedef __attribute__((ext_vector_type(2))) unsigned int v2u;
typedef __attribute__((ext_vector_type(4))) float        v4f;

// ---------------------------------------------------------------------------
// Zero the output (harness poisons d_out with 0xAA before timing).
// 262144 floats = 65536 float4 stores.
// ---------------------------------------------------------------------------
__global__ __launch_bounds__(256) void vdw_zero_kernel(float* __restrict__ out, int n4) {
    int i = blockIdx.x * blockDim.x + threadIdx.x;
    if (i < n4) {
        ((v4f*)out)[i] = (v4f){0.0f, 0.0f, 0.0f, 0.0f};
    }
}

// ---------------------------------------------------------------------------
// Main scatter kernel: one atom per lane (wave32 -> 32 atoms/wave).
//
//   at,b,c,r  = atom_description[i, 0..3]          (16B NT load)
//   mask      = alternativeMask[i, 0..7]           (8 bool bytes, 8B NT load)
//   f         = facc[i, 0..7]                      (2x 16B NT load)
//   vdw       = atom_Properties[at, 0]             (gather; 40 hot rows -> L0)
//   scale     = (1 - tanh(weight[0])) * 0.3        (uniform -> scalar load)
//   val[k]    = mask[k] ? vdw * max(f[k],0) * scale : skip
//   dst       = (at<=3 ? MC : SC) + ((b*4 + c)*512 + r)*8
//   atomicAdd(dst+k, val[k])   -- predicated per lane: masked-off lanes
//                                 generate no L2 atomic traffic.
// ---------------------------------------------------------------------------
__global__ __launch_bounds__(256) void vdw_scatter_kernel(
    const int*           __restrict__ ad,      // (N,4) int32
    const unsigned char* __restrict__ amask,   // (N,8) bool bytes
    const float*         __restrict__ facc,    // (N,8) f32
    const float*         __restrict__ props,   // (500,8) f32
    const float*         __restrict__ weight,  // (1,)  f32
    float*               __restrict__ out,     // [2 * OUT_PER] f32
    int n)
{
    int i = blockIdx.x * blockDim.x + threadIdx.x;
    if (i >= n) return;

    // Uniform address -> scalar load; tanh once per lane (VALU, negligible).
    const float scale = (1.0f - tanhf(weight[0])) * 0.3f;

    // Streaming 128-bit non-temporal loads (single-pass data, keep L2 for output).
    const v4i d  = __builtin_nontemporal_load((const v4i*)ad + i);
    const v2u mw = __builtin_nontemporal_load((const v2u*)amask + i);
    const v4f f0 = __builtin_nontemporal_load((const v4f*)facc + 2 * i);
    const v4f f1 = __builtin_nontemporal_load((const v4f*)facc + 2 * i + 1);

    const int at = d.x, b = d.y, c = d.z, r = d.w;

    if (r == -1) return;  // PADDING_INDEX

    // vdw gather (VDW_COL = 0), fold in the uniform scale.
    const float vdw = props[at * 8] * scale;

    // BACKBONE_ATOMS = {0,1,2,3}; at in [0,40)
    const bool is_bb = ((unsigned)at) <= 3u;

    float* dst = out + (is_bb ? 0 : OUT_PER)
                     + ((((b * C_DIM) + c) * R_DIM + r) << 3);

    const float v0 = vdw * fmaxf(f0.x, 0.0f);
    const float v1 = vdw * fmaxf(f0.y, 0.0f);
    const float v2 = vdw * fmaxf(f0.z, 0.0f);
    const float v3 = vdw * fmaxf(f0.w, 0.0f);
    const float v4 = vdw * fmaxf(f1.x, 0.0f);
    const float v5 = vdw * fmaxf(f1.y, 0.0f);
    const float v6 = vdw * fmaxf(f1.z, 0.0f);
    const float v7 = vdw * fmaxf(f1.w, 0.0f);

    // Per-alt predication: ~50% of mask bytes are 0 -> halves L2 atomic traffic.
    if (mw.x & 0x000000FFu) atomicAdd(dst + 0, v0);
    if (mw.x & 0x0000FF00u) atomicAdd(dst + 1, v1);
    if (mw.x & 0x00FF0000u) atomicAdd(dst + 2, v2);
    if (mw.x & 0xFF000000u) atomicAdd(dst + 3, v3);
    if (mw.y & 0x000000FFu) atomicAdd(dst + 4, v4);
    if (mw.y & 0x0000FF00u) atomicAdd(dst + 5, v5);
    if (mw.y & 0x00FF0000u) atomicAdd(dst + 6, v6);
    if (mw.y & 0xFF000000u) atomicAdd(dst + 7, v7);
}

// ---------------------------------------------------------------------------
// Inputs (setup_inputs order):
//   d_in[0] coords            (N,3)   f32   -- UNUSED by reference
//   d_in[1] atom_description  (N,4)   int
//   d_in[2] alternativeMask   (N,8)   bool
//   d_in[3] facc              (N,8)   f32
//   d_in[4] atom_Properties   (500,8) f32
//   d_in[5] weight            (1,)    f32
// Output: finalMC (8,4,512,8) then finalSC (8,4,512,8), flat f32.
// ---------------------------------------------------------------------------
extern "C" void kernel_launch(void* const* d_in, const int* in_sizes, int n_in,
                              void* d_out, int out_size, void* d_ws, size_t ws_size,
                              hipStream_t stream)
{
    const int*           ad     = (const int*)d_in[1];
    const unsigned char* amask  = (const unsigned char*)d_in[2];
    const float*         facc   = (const float*)d_in[3];
    const float*         props  = (const float*)d_in[4];
    const float*         weight = (const float*)d_in[5];
    float*               out    = (float*)d_out;

    const int n  = in_sizes[1] / 4;     // number of atoms
    const int n4 = out_size / 4;        // float4 count for zeroing

    vdw_zero_kernel<<<(n4 + 255) / 256, 256, 0, stream>>>(out, n4);
    vdw_scatter_kernel<<<(n + 255) / 256, 256, 0, stream>>>(
        ad, amask, facc, props, weight, out, n);
}